// ComplexBatchNorm2d_36807869727213
// MI455X (gfx1250) — compile-verified
//
#include <hip/hip_runtime.h>

// Problem constants (fixed by the reference):  [B=8, C=64, H=256, W=256, 2]
#define NB    8
#define NC    64
#define NHW   65536       // H*W
#define HW2   131072      // H*W*2 floats per (b,c) slice
#define HW2V  32768       // float4 count per (b,c) slice
#define NPART 64          // partition blocks per channel
#define NTH   256
#define EPSV  1e-5f

typedef __attribute__((ext_vector_type(2))) float v2f;
typedef __attribute__((ext_vector_type(4))) float v4f;   // 16B, b128-capable
typedef __attribute__((ext_vector_type(8))) float v8f;

// ---------------------------------------------------------------------------
// Exact f32 wave32 sum via V_WMMA_F32_16X16X4_F32.
// A (16x4, 2 VGPRs/lane): lanes 0-15 hold K=0 (a[0]) / K=1 (a[1]);
// lanes 16-31 hold K=2 / K=3.  With a[1]=0 and B = ones(4x16):
//   D[m][n] = x[m] + x[m+16]   (replicated over n)
// D layout: vgpr j = row (j or j+8).  Sum the 8 D regs per lane, then add
// the opposite half with shfl_xor(16) -> every lane holds the full sum.
// ---------------------------------------------------------------------------
__device__ __forceinline__ float wave_reduce_sum_wmma(float x) {
  v2f a; a[0] = x;    a[1] = 0.0f;
  v2f b; b[0] = 1.0f; b[1] = 1.0f;
  v8f c = {};
  c = __builtin_amdgcn_wmma_f32_16x16x4_f32(
      /*neg_a=*/false, a, /*neg_b=*/false, b,
      /*c_mod=*/(short)0, c, /*reuse_a=*/false, /*reuse_b=*/false);
  float t = c[0] + c[1] + c[2] + c[3] + c[4] + c[5] + c[6] + c[7];
  t += __shfl_xor(t, 16, 32);
  return t;
}

// ---------------------------------------------------------------------------
// Kernel 1: per-(channel, partition) raw moments.
// part[(c*NPART + p)*5 + {0..4}] = {Sr, Si, Srr, Sri, Sii}
// ---------------------------------------------------------------------------
__global__ void cbn_stats_kernel(const float* __restrict__ x,
                                 float* __restrict__ part) {
  const int c   = blockIdx.x;
  const int p   = blockIdx.y;
  const int tid = threadIdx.x;
  const v4f* xv = (const v4f*)x;

  float sr = 0.f, si = 0.f, srr = 0.f, sri = 0.f, sii = 0.f;
  for (int b = 0; b < NB; ++b) {
    const size_t base = (size_t)(b * NC + c) * HW2V;
    // HW2V = 32768, stride NPART*NTH = 16384 -> exactly 2 uniform iterations
    for (int i = p * NTH + tid; i < HW2V; i += NPART * NTH) {
      v4f v = __builtin_nontemporal_load(&xv[base + i]);  // (r0,i0,r1,i1)
      sr  += v[0] + v[2];
      si  += v[1] + v[3];
      srr += v[0] * v[0] + v[2] * v[2];
      sri += v[0] * v[1] + v[2] * v[3];
      sii += v[1] * v[1] + v[3] * v[3];
    }
  }

  // wave32 reduce via WMMA (EXEC all ones here)
  sr  = wave_reduce_sum_wmma(sr);
  si  = wave_reduce_sum_wmma(si);
  srr = wave_reduce_sum_wmma(srr);
  sri = wave_reduce_sum_wmma(sri);
  sii = wave_reduce_sum_wmma(sii);

  __shared__ float lpart[NTH / 32][5];
  const int wave = tid >> 5, lane = tid & 31;
  if (lane == 0) {
    lpart[wave][0] = sr;  lpart[wave][1] = si;
    lpart[wave][2] = srr; lpart[wave][3] = sri; lpart[wave][4] = sii;
  }
  __syncthreads();
  if (tid < 5) {
    float s = 0.f;
    for (int w = 0; w < NTH / 32; ++w) s += lpart[w][tid];
    part[((size_t)c * NPART + p) * 5 + tid] = s;
  }
}

// ---------------------------------------------------------------------------
// Kernel 2: one wave per channel. Reduce partials, whiten, fold gamma/beta/mu
// into 6 coefficients: out_r = arr*r + ari*i + br ; out_i = air*r + aii*i + bi
// ---------------------------------------------------------------------------
__global__ void cbn_finalize_kernel(const float* __restrict__ part,
                                    const float* __restrict__ g_rr,
                                    const float* __restrict__ g_ii,
                                    const float* __restrict__ g_ri,
                                    const float* __restrict__ b_r,
                                    const float* __restrict__ b_i,
                                    float* __restrict__ coef) {
  const int c    = blockIdx.x;
  const int lane = threadIdx.x;  // 32 threads

  float sr = 0.f, si = 0.f, srr = 0.f, sri = 0.f, sii = 0.f;
  for (int p = lane; p < NPART; p += 32) {          // NPART=64 -> uniform
    const float* q = &part[((size_t)c * NPART + p) * 5];
    sr += q[0]; si += q[1]; srr += q[2]; sri += q[3]; sii += q[4];
  }
  sr  = wave_reduce_sum_wmma(sr);
  si  = wave_reduce_sum_wmma(si);
  srr = wave_reduce_sum_wmma(srr);
  sri = wave_reduce_sum_wmma(sri);
  sii = wave_reduce_sum_wmma(sii);

  if (lane == 0) {
    const float invN = 1.0f / ((float)NB * (float)NHW);
    const float mur = sr * invN, mui = si * invN;
    const float Vrr = srr * invN - mur * mur + EPSV;
    const float Vri = sri * invN - mur * mui + EPSV;
    const float Vii = sii * invN - mui * mui + EPSV;
    const float tau = Vrr + Vii;
    const float dlt = Vrr * Vii - Vri * Vri;
    const float s   = sqrtf(dlt);
    const float t   = sqrtf(tau + 2.0f * s);
    const float inv = 1.0f / (s * t);
    const float Wrr = (Vii + s) * inv;
    const float Wri = -Vri * inv;
    const float Wii = (Vrr + s) * inv;

    const float grr = g_rr[c], gii = g_ii[c], gri = g_ri[c];
    const float br0 = b_r[c],  bi0 = b_i[c];

    const float arr = grr * Wrr + gri * Wri;
    const float ari = grr * Wri + gri * Wii;
    const float air = gri * Wrr + gii * Wri;
    const float aii = gri * Wri + gii * Wii;

    float* q = &coef[c * 8];
    q[0] = arr; q[1] = ari; q[2] = br0 - arr * mur - ari * mui;
    q[3] = air; q[4] = aii; q[5] = bi0 - air * mur - aii * mui;
  }
}

// ---------------------------------------------------------------------------
// Kernel 3: streaming apply, b128 NT load -> 4 FMAs -> b128 NT store.
// ---------------------------------------------------------------------------
__global__ void cbn_apply_kernel(const float* __restrict__ x,
                                 const float* __restrict__ coef,
                                 float* __restrict__ out) {
  const int c   = blockIdx.x;
  const int p   = blockIdx.y;
  const int tid = threadIdx.x;

  const float* q = &coef[c * 8];
  const float arr = q[0], ari = q[1], br = q[2];
  const float air = q[3], aii = q[4], bi = q[5];

  const v4f* xv = (const v4f*)x;
  v4f*       ov = (v4f*)out;

  for (int b = 0; b < NB; ++b) {
    const size_t base = (size_t)(b * NC + c) * HW2V;
    for (int i = p * NTH + tid; i < HW2V; i += NPART * NTH) {
      v4f v = __builtin_nontemporal_load(&xv[base + i]);
      v4f o;
      o[0] = arr * v[0] + ari * v[1] + br;
      o[1] = air * v[0] + aii * v[1] + bi;
      o[2] = arr * v[2] + ari * v[3] + br;
      o[3] = air * v[2] + aii * v[3] + bi;
      __builtin_nontemporal_store(o, &ov[base + i]);
    }
  }
}

// ---------------------------------------------------------------------------
extern "C" void kernel_launch(void* const* d_in, const int* in_sizes, int n_in,
                              void* d_out, int out_size, void* d_ws, size_t ws_size,
                              hipStream_t stream) {
  (void)in_sizes; (void)n_in; (void)out_size; (void)ws_size;

  const float* x    = (const float*)d_in[0];
  const float* g_rr = (const float*)d_in[1];
  const float* g_ii = (const float*)d_in[2];
  const float* g_ri = (const float*)d_in[3];
  const float* b_r  = (const float*)d_in[4];
  const float* b_i  = (const float*)d_in[5];
  float*       out  = (float*)d_out;

  // Workspace layout: partials (C*NPART*5 = 20480 f), coefficients (C*8 f)
  float* part = (float*)d_ws;
  float* coef = part + (size_t)NC * NPART * 5;

  cbn_stats_kernel<<<dim3(NC, NPART), NTH, 0, stream>>>(x, part);
  cbn_finalize_kernel<<<NC, 32, 0, stream>>>(part, g_rr, g_ii, g_ri, b_r, b_i, coef);
  cbn_apply_kernel<<<dim3(NC, NPART), NTH, 0, stream>>>(x, coef, out);
}